// GCNAggregator_26963804685001
// MI455X (gfx1250) — compile-verified
//
#include <hip/hip_runtime.h>

typedef float v2f __attribute__((ext_vector_type(2)));
typedef float v4f __attribute__((ext_vector_type(4)));
typedef float v8f __attribute__((ext_vector_type(8)));

#define NROWS 16384
#define KNEIGH 25
#define DIM 512
#define ODIM 1024

// ---------------------------------------------------------------------------
// Kernel 1: means[n,d] = (sum_k neigh[n,k,d] + self[n,d]) / (K+1)
// Pure streaming, v4f vectorized, non-temporal loads on the 800MB input.
// ---------------------------------------------------------------------------
__global__ __launch_bounds__(256) void gcn_mean_kernel(
    const float* __restrict__ self_vecs,
    const float* __restrict__ neigh,
    float* __restrict__ means)
{
    const int idx = blockIdx.x * 256 + threadIdx.x;   // v4f index, exact cover
    const int n   = idx >> 7;                         // / (DIM/4 = 128)
    const int d4  = idx & 127;

    const v4f* self4 = reinterpret_cast<const v4f*>(self_vecs);
    const v4f* nb4   = reinterpret_cast<const v4f*>(neigh);

    v4f acc = self4[n * 128 + d4];
    const v4f* p = nb4 + (size_t)n * (KNEIGH * 128) + d4;
#pragma unroll
    for (int k = 0; k < KNEIGH; ++k) {
        v4f v = __builtin_nontemporal_load(p + k * 128);
        acc += v;
    }
    acc *= (1.0f / (float)(KNEIGH + 1));
    reinterpret_cast<v4f*>(means)[n * 128 + d4] = acc;
}

// ---------------------------------------------------------------------------
// Kernel 2: out = relu(means @ W + b) using V_WMMA_F32_16X16X4_F32.
// Block: 256 threads (8 waves). Block computes 16 rows x 1024 cols.
// Wave w owns cols [w*128, w*128+128) as 8 accumulator tiles of 16x16.
// A tile (16x512 of means, 32KB) staged in LDS, shared by all waves.
// W streamed from global (2MB total, L2-resident across all 1024 blocks).
// ---------------------------------------------------------------------------
__global__ __launch_bounds__(256) void gcn_gemm_kernel(
    const float* __restrict__ means,
    const float* __restrict__ W,
    const float* __restrict__ bias,
    float* __restrict__ out)
{
    __shared__ float smeans[16 * DIM];   // 32 KB

    const int tid  = threadIdx.x;
    const int row0 = blockIdx.x * 16;

    // Stage the 16x512 means tile into LDS (2048 v4fs / 256 threads = 8 each)
    const v4f* m4 = reinterpret_cast<const v4f*>(means) + (size_t)row0 * (DIM / 4);
    v4f* s4 = reinterpret_cast<v4f*>(smeans);
#pragma unroll
    for (int i = 0; i < 8; ++i) {
        int f = tid + i * 256;
        s4[f] = m4[f];
    }
    __syncthreads();

    const int wave    = tid >> 5;     // wave32
    const int lane    = tid & 31;
    const int hi      = lane >> 4;    // upper/lower 16-lane half
    const int nl      = lane & 15;
    const int colbase = wave * 128;

    v8f acc[8];
#pragma unroll
    for (int t = 0; t < 8; ++t) acc[t] = (v8f)0.0f;

    const float* srow = smeans + nl * DIM;   // A row for this lane (m = nl)

    for (int ko = 0; ko < DIM; ko += 4) {
        const int k0 = ko + 2 * hi;          // A/B VGPR0 K-index for this half
        v2f a;
        a.x = srow[k0];                      // A[m, k0]
        a.y = srow[k0 + 1];                  // A[m, k0+1]
        const float* w0 = W + (size_t)k0 * ODIM + nl;
#pragma unroll
        for (int t = 0; t < 8; ++t) {
            const int c0 = colbase + t * 16;
            v2f b;
            b.x = w0[c0];                    // B[k0,   n]
            b.y = w0[ODIM + c0];             // B[k0+1, n]
            acc[t] = __builtin_amdgcn_wmma_f32_16x16x4_f32(
                false, a, false, b, (short)0, acc[t], false, false);
        }
    }

    // Epilogue: bias + relu + store. D layout: row = v + 8*hi, col = nl.
#pragma unroll
    for (int t = 0; t < 8; ++t) {
        const int c  = colbase + t * 16 + nl;
        const float bv = bias[c];
#pragma unroll
        for (int v = 0; v < 8; ++v) {
            const int m = v + 8 * hi;
            float val = acc[t][v] + bv;
            val = val > 0.0f ? val : 0.0f;
            out[(size_t)(row0 + m) * ODIM + c] = val;
        }
    }
}

// ---------------------------------------------------------------------------
// Launch: inputs are {self_vecs, neigh_vecs, temperature(unused), W, b}
// ---------------------------------------------------------------------------
extern "C" void kernel_launch(void* const* d_in, const int* in_sizes, int n_in,
                              void* d_out, int out_size, void* d_ws, size_t ws_size,
                              hipStream_t stream) {
    const float* self_vecs = (const float*)d_in[0];
    const float* neigh     = (const float*)d_in[1];
    const float* W         = (const float*)d_in[3];
    const float* b         = (const float*)d_in[4];
    float* out   = (float*)d_out;
    float* means = (float*)d_ws;   // needs 16384*512*4 = 32 MB scratch

    // Stage 1: means (N*D/4 v4f elements, 256 threads/block)
    const int nThreads1 = (NROWS * DIM) / 4;           // 2,097,152
    gcn_mean_kernel<<<nThreads1 / 256, 256, 0, stream>>>(self_vecs, neigh, means);

    // Stage 2: WMMA GEMM, one block per 16 output rows
    gcn_gemm_kernel<<<NROWS / 16, 256, 0, stream>>>(means, W, b, out);
}